// AutoF1LSTM_78786880078505
// MI455X (gfx1250) — compile-verified
//
#include <hip/hip_runtime.h>
#include <hip/hip_bf16.h>
#include <math.h>

typedef float v2f __attribute__((ext_vector_type(2)));
typedef float v8f __attribute__((ext_vector_type(8)));

#define H 2048
#define LIN 92

// ---------------------------------------------------------------------------
// Build x[92] = [lap_cont(60), team_e(8), trk_e(8), drv_e(8), comp_e(8)]
// idx layout: [compound, driver, track, team]
// ---------------------------------------------------------------------------
__global__ void build_x_kernel(const float* __restrict__ lap,
                               const int* __restrict__ idx,
                               const float* __restrict__ e_team,
                               const float* __restrict__ e_track,
                               const float* __restrict__ e_driver,
                               const float* __restrict__ e_comp,
                               float* __restrict__ x) {
    int t = threadIdx.x;
    if (t < 60)       x[t] = lap[t];
    else if (t < 68)  x[t] = e_team  [idx[3] * 8 + (t - 60)];
    else if (t < 76)  x[t] = e_track [idx[2] * 8 + (t - 68)];
    else if (t < 84)  x[t] = e_driver[idx[1] * 8 + (t - 76)];
    else if (t < 92)  x[t] = e_comp  [idx[0] * 8 + (t - 84)];
}

// ---------------------------------------------------------------------------
// WMMA dual GEMV:  y[j] = W1[j,:]·v1 + W2[j,:]·v2 + b1[j] + b2[j]
//
// One wave computes 32 output rows with two v8f accumulators sharing one
// B-operand fetch. A tiles are 16x4 fp32 weight blocks loaded as per-lane
// b64 (each weight byte read exactly once, global path carries only the
// weight stream). The broadcast vector is staged in LDS per block, so the
// inner loop is: 2x global_load_b64 + 1x ds_load_b64 + 2x v_wmma.
// Grids are sized so every wave is full (WMMA requires EXEC all-1s).
// ---------------------------------------------------------------------------
__global__ void gemv_wmma_kernel(const float* __restrict__ W1,
                                 const float* __restrict__ v1, int K1,
                                 const float* __restrict__ W2,
                                 const float* __restrict__ v2, int K2,
                                 const float* __restrict__ b1,
                                 const float* __restrict__ b2,
                                 float* __restrict__ y,
                                 int do_relu) {
    __shared__ float sv[2 * H];          // staged v1 | v2 (16 KB of 320 KB)

    // cooperative stage of the broadcast operands
    for (int j = threadIdx.x; j < K1; j += blockDim.x) sv[j] = v1[j];
    if (K2 > 0)
        for (int j = threadIdx.x; j < K2; j += blockDim.x) sv[H + j] = v2[j];
    __syncthreads();

    const int lane = threadIdx.x & 31;
    const int wave = blockIdx.x * (blockDim.x >> 5) + (threadIdx.x >> 5);
    const int row0 = wave * 32;          // 32 rows per wave; grid keeps valid
    const int mrow = row0 + (lane & 15); // lane's weight row (first tile)
    const int hi   = lane >> 4;          // half-wave K-pair selector

    v8f acc0 = {0.f, 0.f, 0.f, 0.f, 0.f, 0.f, 0.f, 0.f};
    v8f acc1 = {0.f, 0.f, 0.f, 0.f, 0.f, 0.f, 0.f, 0.f};

    // --- matrix 1 ---
    {
        const float* wra = W1 + (size_t)mrow * (size_t)K1;
        const float* wrb = wra + (size_t)16 * (size_t)K1;
        #pragma unroll 4
        for (int k = 0; k < K1; k += 4) {
            const int c = k + 2 * hi;
            v2f a0 = *(const v2f*)(wra + c);       // rows row0..row0+15
            v2f a1 = *(const v2f*)(wrb + c);       // rows row0+16..row0+31
            v2f b  = *(const v2f*)(sv + c);        // LDS broadcast operand
            acc0 = __builtin_amdgcn_wmma_f32_16x16x4_f32(
                false, a0, false, b, (short)0, acc0, false, false);
            acc1 = __builtin_amdgcn_wmma_f32_16x16x4_f32(
                false, a1, false, b, (short)0, acc1, false, false);
        }
    }
    // --- matrix 2 (optional) ---
    if (K2 > 0) {
        const float* wra = W2 + (size_t)mrow * (size_t)K2;
        const float* wrb = wra + (size_t)16 * (size_t)K2;
        #pragma unroll 4
        for (int k = 0; k < K2; k += 4) {
            const int c = k + 2 * hi;
            v2f a0 = *(const v2f*)(wra + c);
            v2f a1 = *(const v2f*)(wrb + c);
            v2f b  = *(const v2f*)(sv + H + c);
            acc0 = __builtin_amdgcn_wmma_f32_16x16x4_f32(
                false, a0, false, b, (short)0, acc0, false, false);
            acc1 = __builtin_amdgcn_wmma_f32_16x16x4_f32(
                false, a1, false, b, (short)0, acc1, false, false);
        }
    }

    // D layout: VGPR r, lanes 0-15 -> M=r, lanes 16-31 -> M=8+r (all N equal)
    if ((lane & 15) == 0) {
        const int base = row0 + 8 * hi;
        #pragma unroll
        for (int r = 0; r < 8; ++r) {
            float val = acc0[r];
            const int j = base + r;
            if (b1) val += b1[j];
            if (b2) val += b2[j];
            if (do_relu) val = fmaxf(val, 0.f);
            y[j] = val;
        }
        #pragma unroll
        for (int r = 0; r < 8; ++r) {
            float val = acc1[r];
            const int j = base + 16 + r;
            if (b1) val += b1[j];
            if (b2) val += b2[j];
            if (do_relu) val = fmaxf(val, 0.f);
            y[j] = val;
        }
    }
}

// ---------------------------------------------------------------------------
// LSTM cell nonlinearity: gates g[0..8191] = [i, f, g, o]
// ---------------------------------------------------------------------------
__device__ __forceinline__ float sigmoid_f(float x) {
    return 1.f / (1.f + expf(-x));
}

__global__ void cell_update_kernel(const float* __restrict__ g,
                                   const float* __restrict__ c_in,
                                   float* __restrict__ h_ws,
                                   float* __restrict__ h_out,
                                   float* __restrict__ c_out) {
    int j = blockIdx.x * blockDim.x + threadIdx.x;
    if (j >= H) return;
    float gi = g[j];
    float gf = g[H + j];
    float gg = g[2 * H + j];
    float go = g[3 * H + j];
    float c  = sigmoid_f(gf) * c_in[j] + sigmoid_f(gi) * tanhf(gg);
    float h  = sigmoid_f(go) * tanhf(c);
    c_out[j] = c;
    h_out[j] = h;
    h_ws[j]  = h;
}

// ---------------------------------------------------------------------------
// LayerNorm over 2048 elements, single block of 256 threads
// ---------------------------------------------------------------------------
__global__ void layernorm_kernel(const float* __restrict__ h,
                                 const float* __restrict__ gam,
                                 const float* __restrict__ bet,
                                 float* __restrict__ out) {
    __shared__ float sm[256];
    const int t = threadIdx.x;

    float s = 0.f;
    for (int j = t; j < H; j += 256) s += h[j];
    sm[t] = s; __syncthreads();
    for (int off = 128; off > 0; off >>= 1) {
        if (t < off) sm[t] += sm[t + off];
        __syncthreads();
    }
    const float mu = sm[0] * (1.f / H);
    __syncthreads();

    float v = 0.f;
    for (int j = t; j < H; j += 256) { float d = h[j] - mu; v += d * d; }
    sm[t] = v; __syncthreads();
    for (int off = 128; off > 0; off >>= 1) {
        if (t < off) sm[t] += sm[t + off];
        __syncthreads();
    }
    const float var = sm[0] * (1.f / H);
    const float r = rsqrtf(var + 1e-5f);

    for (int j = t; j < H; j += 256)
        out[j] = (h[j] - mu) * r * gam[j] + bet[j];
}

// ---------------------------------------------------------------------------
// Head stage 2: outs[k][p] = head_W2[k][p]·hid[k] + head_b2[k][p]; write only
// the HEAD_OUTS[k] used lanes, packed into 27 floats.
// ---------------------------------------------------------------------------
__global__ void heads2_kernel(const float* __restrict__ hid,
                              const float* __restrict__ W2,
                              const float* __restrict__ b2,
                              float* __restrict__ out) {
    const int outs[13] = {5, 1, 1, 1, 1, 1, 1, 1, 9, 1, 3, 1, 1};
    const int pre[13]  = {0, 5, 6, 7, 8, 9, 10, 11, 12, 21, 22, 25, 26};
    int t = threadIdx.x;                 // 0..116 used
    if (t >= 117) return;
    int k = t / 9, p = t % 9;
    if (p >= outs[k]) return;
    const float* w  = W2 + ((size_t)k * 9 + p) * 64;
    const float* hv = hid + k * 64;
    float s = b2[k * 9 + p];
    #pragma unroll
    for (int o = 0; o < 64; ++o) s += w[o] * hv[o];
    out[pre[k] + p] = s;
}

// ---------------------------------------------------------------------------
// Launch: x -> L0 gates -> L0 cell -> L1 gates -> L1 cell -> LN -> heads
// ---------------------------------------------------------------------------
extern "C" void kernel_launch(void* const* d_in, const int* in_sizes, int n_in,
                              void* d_out, int out_size, void* d_ws, size_t ws_size,
                              hipStream_t stream) {
    const float* lap      = (const float*)d_in[0];
    const int*   idx      = (const int*)  d_in[1];
    const float* h_s      = (const float*)d_in[2];   // (2,1,H)
    const float* c_s      = (const float*)d_in[3];   // (2,1,H)
    const float* e_team   = (const float*)d_in[4];
    const float* e_track  = (const float*)d_in[5];
    const float* e_driver = (const float*)d_in[6];
    const float* e_comp   = (const float*)d_in[7];
    const float* W_ih0    = (const float*)d_in[8];
    const float* W_hh0    = (const float*)d_in[9];
    const float* b_ih0    = (const float*)d_in[10];
    const float* b_hh0    = (const float*)d_in[11];
    const float* W_ih1    = (const float*)d_in[12];
    const float* W_hh1    = (const float*)d_in[13];
    const float* b_ih1    = (const float*)d_in[14];
    const float* b_hh1    = (const float*)d_in[15];
    const float* ln_g     = (const float*)d_in[16];
    const float* ln_b     = (const float*)d_in[17];
    const float* head_W1  = (const float*)d_in[18];  // (13,64,H) = 832 x H
    const float* head_b1  = (const float*)d_in[19];  // 832
    const float* head_W2  = (const float*)d_in[20];  // (13,9,64)
    const float* head_b2  = (const float*)d_in[21];  // (13,9)

    float* out = (float*)d_out;
    // out layout: h0[0:2048] h1[2048:4096] c0[4096:6144] c1[6144:8192] heads[8192:8219]
    float* out_h0 = out;
    float* out_h1 = out + H;
    float* out_c0 = out + 2 * H;
    float* out_c1 = out + 3 * H;
    float* out_hd = out + 4 * H;

    float* ws  = (float*)d_ws;
    float* x   = ws;              // 92   (16B-aligned, even-stride v2f loads)
    float* g0  = ws + 128;        // 8192
    float* h0  = ws + 8448;       // 2048
    float* g1  = ws + 10496;      // 8192
    float* h1  = ws + 18688;      // 2048
    float* ht  = ws + 20736;      // 2048 (normalized)
    float* hid = ws + 22784;      // 832

    // 1. assemble input vector
    build_x_kernel<<<1, 128, 0, stream>>>(lap, idx, e_team, e_track, e_driver,
                                          e_comp, x);

    // 2. layer-0 gates: 8192 rows / 32 per wave = 256 waves (32 blk x 8 wv)
    gemv_wmma_kernel<<<32, 256, 0, stream>>>(W_ih0, x, LIN,
                                             W_hh0, h_s, H,
                                             b_ih0, b_hh0, g0, 0);
    // 3. layer-0 cell
    cell_update_kernel<<<8, 256, 0, stream>>>(g0, c_s, h0, out_h0, out_c0);

    // 4. layer-1 gates
    gemv_wmma_kernel<<<32, 256, 0, stream>>>(W_ih1, h0, H,
                                             W_hh1, h_s + H, H,
                                             b_ih1, b_hh1, g1, 0);
    // 5. layer-1 cell
    cell_update_kernel<<<8, 256, 0, stream>>>(g1, c_s + H, h1, out_h1, out_c1);

    // 6. layernorm
    layernorm_kernel<<<1, 256, 0, stream>>>(h1, ln_g, ln_b, ht);

    // 7. head stage 1: 832 rows / 32 per wave = 26 waves (13 blk x 2 wv), relu
    gemv_wmma_kernel<<<13, 64, 0, stream>>>(head_W1, ht, H,
                                            nullptr, nullptr, 0,
                                            head_b1, nullptr, hid, 1);

    // 8. head stage 2: 27 packed outputs
    heads2_kernel<<<1, 128, 0, stream>>>(hid, head_W2, head_b2, out_hd);
}